// HistogramLoss_88433376625133
// MI455X (gfx1250) — compile-verified
//
#include <hip/hip_runtime.h>
#include <hip/hip_bf16.h>

#define NUM_BINS 256
#define BC       48            // B*C = 16*3
#define IMG_ELEMS (512 * 512)  // elements per (b,c) image
#define CHUNKS   16            // blocks per image per tensor
#define THREADS  256           // 8 wave32 per block
#define HEPS     1e-8f

typedef __attribute__((ext_vector_type(2))) float v2f;
typedef __attribute__((ext_vector_type(8))) float v8f;

// ---------------------------------------------------------------------------
// Kernel 1: zero the global histogram workspace (d_ws is poisoned to 0xAA).
// ---------------------------------------------------------------------------
__global__ __launch_bounds__(THREADS)
void hl_zero_kernel(unsigned int* __restrict__ ws, int n) {
  int i = blockIdx.x * THREADS + threadIdx.x;
  if (i < n) ws[i] = 0u;
}

// ---------------------------------------------------------------------------
// Kernel 2: streaming histogram. Each block handles one (tensor, bc, chunk):
// 16384 f32 elements via float4 loads -> per-wave LDS histograms (ds_add),
// then one global atomicAdd per bin. torch.histc semantics:
// bin = clip(floor(x*256), 0, 255); values outside [0,1] ignored.
// ---------------------------------------------------------------------------
__global__ __launch_bounds__(THREADS)
void hl_hist_kernel(const float* __restrict__ pred,
                    const float* __restrict__ target,
                    unsigned int* __restrict__ ghist) {
  __shared__ unsigned int lhist[8 * NUM_BINS];  // one 256-bin hist per wave32
  const int tid = threadIdx.x;
  const int wid = tid >> 5;

  #pragma unroll
  for (int i = tid; i < 8 * NUM_BINS; i += THREADS) lhist[i] = 0u;
  __syncthreads();

  const int blk    = blockIdx.x;               // [0, 2*BC*CHUNKS)
  const int tensor = blk / (BC * CHUNKS);
  const int rem    = blk % (BC * CHUNKS);
  const int bc     = rem / CHUNKS;
  const int chunk  = rem % CHUNKS;
  const float* src = tensor ? target : pred;

  const int chunk_elems = IMG_ELEMS / CHUNKS;  // 16384
  const float4* q = (const float4*)(src + (size_t)bc * IMG_ELEMS
                                        + (size_t)chunk * chunk_elems) + tid;
  unsigned int* myhist = &lhist[wid * NUM_BINS];

  const int iters = chunk_elems / 4 / THREADS; // 16 float4 per thread
  for (int it = 0; it < iters; ++it) {
    if (it + 2 < iters)  // gfx1250 global_prefetch_b8, 2 iterations ahead
      __builtin_prefetch((const void*)(q + (size_t)(it + 2) * THREADS), 0, 1);
    float4 v = q[(size_t)it * THREADS];
    float e[4] = { v.x, v.y, v.z, v.w };
    #pragma unroll
    for (int j = 0; j < 4; ++j) {
      float x = e[j];
      int b = (int)(x * 256.0f);        // trunc == floor for x >= 0 (exact *2^8)
      b = b > 255 ? 255 : b;            // x == 1.0 -> last bin
      if (x >= 0.0f && x <= 1.0f) atomicAdd(&myhist[b], 1u);
    }
  }
  __syncthreads();

  unsigned int* gout = &ghist[((size_t)tensor * BC + bc) * NUM_BINS];
  for (int t = tid; t < NUM_BINS; t += THREADS) {
    unsigned int s = 0;
    #pragma unroll
    for (int w = 0; w < 8; ++w) s += lhist[w * NUM_BINS + t];
    if (s) atomicAdd(&gout[t], s);
  }
}

// ---------------------------------------------------------------------------
// Kernel 3: finalize (1 wave32, EXEC all-ones as WMMA requires).
// Step 1: per-histogram sums Sp[48], St[48] as a [48x256] x ones[256] GEMV
//         with V_WMMA_F32_16X16X4_F32 (f32 keeps integer counts exact).
//         A-tile layout (16x4 f32): lanes 0-15 carry K=0,1; lanes 16-31 K=2,3.
//         C/D layout: VGPR r = rows {r, r+8}, row-sum replicated across N.
// Step 2: loss = mean over 12288 bins of (p/Sp - t/St)^2, double accumulate.
// ---------------------------------------------------------------------------
__global__ __launch_bounds__(32)
void hl_final_kernel(const unsigned int* __restrict__ ghist,
                     float* __restrict__ out) {
  __shared__ float sums[2 * BC];   // [tensor][bc]
  const int lane = threadIdx.x;

#if __has_builtin(__builtin_amdgcn_wmma_f32_16x16x4_f32)
  {
    const int half = lane >> 4;    // 0: K+0/K+1, 1: K+2/K+3
    const int m    = lane & 15;    // row within M-tile
    v2f bone; bone[0] = 1.0f; bone[1] = 1.0f;   // B = ones(4x16)
    for (int tensor = 0; tensor < 2; ++tensor) {
      for (int tile = 0; tile < 3; ++tile) {     // 3 x 16 = 48 rows
        const unsigned int* row =
            ghist + ((size_t)tensor * BC + (size_t)(tile * 16 + m)) * NUM_BINS;
        v8f c = {};
        #pragma unroll 4
        for (int k = 0; k < NUM_BINS / 4; ++k) { // 64 K-chunks of 4
          const int kb = 4 * k + 2 * half;
          v2f a;
          a[0] = (float)row[kb];
          a[1] = (float)row[kb + 1];
          c = __builtin_amdgcn_wmma_f32_16x16x4_f32(false, a, false, bone,
                                                    (short)0, c, false, false);
        }
        if (m == 0) {  // lane 0 -> rows tile*16+0..7, lane 16 -> +8..15
          float* s = &sums[tensor * BC + tile * 16 + half * 8];
          #pragma unroll
          for (int r = 0; r < 8; ++r) s[r] = c[r];
        }
      }
    }
  }
#else
  for (int j = lane; j < 2 * BC; j += 32) {
    const unsigned int* row = ghist + (size_t)j * NUM_BINS;
    float s = 0.0f;
    for (int b = 0; b < NUM_BINS; ++b) s += (float)row[b];
    sums[j] = s;
  }
#endif
  __syncthreads();

  double acc = 0.0;
  for (int i = lane; i < BC * NUM_BINS; i += 32) {
    const int bc = i >> 8;
    float p = (float)ghist[i]                 / (sums[bc] + HEPS);
    float t = (float)ghist[BC * NUM_BINS + i] / (sums[BC + bc] + HEPS);
    float d = p - t;
    acc += (double)(d * d);
  }
  #pragma unroll
  for (int off = 16; off > 0; off >>= 1)
    acc += __shfl_down(acc, off, 32);
  if (lane == 0) out[0] = (float)(acc / (double)(BC * NUM_BINS));
}

// ---------------------------------------------------------------------------
extern "C" void kernel_launch(void* const* d_in, const int* in_sizes, int n_in,
                              void* d_out, int out_size, void* d_ws, size_t ws_size,
                              hipStream_t stream) {
  (void)in_sizes; (void)n_in; (void)out_size; (void)ws_size;
  const float* pred   = (const float*)d_in[0];
  const float* target = (const float*)d_in[1];
  unsigned int* ghist = (unsigned int*)d_ws;   // 2*48*256 u32 = 96 KB
  float* out          = (float*)d_out;

  const int nhist = 2 * BC * NUM_BINS;
  hl_zero_kernel<<<(nhist + THREADS - 1) / THREADS, THREADS, 0, stream>>>(ghist, nhist);
  hl_hist_kernel<<<2 * BC * CHUNKS, THREADS, 0, stream>>>(pred, target, ghist);
  hl_final_kernel<<<1, 32, 0, stream>>>(ghist, out);
}